// SparseMaxPooling_74294344286532
// MI455X (gfx1250) — compile-verified
//
#include <hip/hip_runtime.h>
#include <hip/hip_bf16.h>
#include <stdint.h>

// Sparse max pooling: out[o][c] = max_{k<8} features[in_map[8*o+k]][c]
// features: [1M, 96] fp32, in_map: permutation of [0,1M) int32, out: [125000, 96] fp32.
//
// Memory-bound: 384MB gathered reads (each row read exactly once; rows are
// 384B = exactly 3 x 128B cachelines, zero over-fetch) + 48MB writes
// ~= 436MB @ 23.3 TB/s ~= 19us floor. 125k wave32s (one per output row) give
// ample latency hiding for the random gather.
//
// CDNA5 specifics used deliberately:
//  - indices are wave-uniform -> one s_load_b256 (SMEM) per wave
//  - row bases computed on the SALU, gathers issued as an s_clause of 8
//    saddr-form global_load_b128 with th:TH_LOAD_NT (single-use stream,
//    384MB > 192MB L2 -> NT avoids thrashing L2 with dead lines)
//  - explicit s_wait_loadcnt with results threaded through the asm so no
//    consumer can be scheduled before the wait
//  - NT store for the write-once output

typedef float f4 __attribute__((ext_vector_type(4)));

__device__ __forceinline__ f4 vmax4(f4 a, f4 b) {
    f4 r;
    r.x = __builtin_fmaxf(a.x, b.x);
    r.y = __builtin_fmaxf(a.y, b.y);
    r.z = __builtin_fmaxf(a.z, b.z);
    r.w = __builtin_fmaxf(a.w, b.w);
    return r;
}

__global__ __launch_bounds__(256)
void SparseMaxPooling_74294344286532_kernel(const float* __restrict__ features,
                                            const int* __restrict__ in_map,
                                            float* __restrict__ out,
                                            int n_out) {
    // Wave-uniform output row id, forced into an SGPR.
    const int wave = __builtin_amdgcn_readfirstlane((int)(threadIdx.x >> 5));
    const int lane = (int)(threadIdx.x & 31u);
    const int row  = blockIdx.x * 8 + wave;
    if (row >= n_out) return;

    // 8 pooling indices: contiguous + wave-uniform -> one s_load_b256.
    int idx[8];
#pragma unroll
    for (int k = 0; k < 8; ++k)
        idx[k] = in_map[row * 8 + k];

    // Row base addresses on the SALU (uniform): features + idx*384 bytes.
    const uint64_t fb = (uint64_t)(uintptr_t)features;
    const uint64_t b0 = fb + (uint64_t)(uint32_t)idx[0] * 384u;
    const uint64_t b1 = fb + (uint64_t)(uint32_t)idx[1] * 384u;
    const uint64_t b2 = fb + (uint64_t)(uint32_t)idx[2] * 384u;
    const uint64_t b3 = fb + (uint64_t)(uint32_t)idx[3] * 384u;
    const uint64_t b4 = fb + (uint64_t)(uint32_t)idx[4] * 384u;
    const uint64_t b5 = fb + (uint64_t)(uint32_t)idx[5] * 384u;
    const uint64_t b6 = fb + (uint64_t)(uint32_t)idx[6] * 384u;
    const uint64_t b7 = fb + (uint64_t)(uint32_t)idx[7] * 384u;

    // One 96-float row = 384B = 24 lanes x b128 (16B, naturally aligned).
    if (lane < 24) {
        const unsigned voff = (unsigned)lane * 16u;
        f4 v0, v1, v2, v3, v4, v5, v6, v7;
        // Clause of 8 saddr-form NT gathers: scalar 64-bit base per row,
        // shared 32-bit per-lane offset.
        asm volatile(
            "s_clause 0x7\n\t"
            "global_load_b128 %0, %8, %9  th:TH_LOAD_NT\n\t"
            "global_load_b128 %1, %8, %10 th:TH_LOAD_NT\n\t"
            "global_load_b128 %2, %8, %11 th:TH_LOAD_NT\n\t"
            "global_load_b128 %3, %8, %12 th:TH_LOAD_NT\n\t"
            "global_load_b128 %4, %8, %13 th:TH_LOAD_NT\n\t"
            "global_load_b128 %5, %8, %14 th:TH_LOAD_NT\n\t"
            "global_load_b128 %6, %8, %15 th:TH_LOAD_NT\n\t"
            "global_load_b128 %7, %8, %16 th:TH_LOAD_NT"
            : "=&v"(v0), "=&v"(v1), "=&v"(v2), "=&v"(v3),
              "=&v"(v4), "=&v"(v5), "=&v"(v6), "=&v"(v7)
            : "v"(voff),
              "s"(b0), "s"(b1), "s"(b2), "s"(b3),
              "s"(b4), "s"(b5), "s"(b6), "s"(b7)
            : "memory");
        // Thread the results through the wait so no consumer is scheduled
        // before LOADcnt reaches 0.
        asm volatile("s_wait_loadcnt 0x0"
                     : "+v"(v0), "+v"(v1), "+v"(v2), "+v"(v3),
                       "+v"(v4), "+v"(v5), "+v"(v6), "+v"(v7)
                     :
                     : "memory");

        // 3-level max tree (v_max_num_f32).
        f4 m = vmax4(vmax4(vmax4(v0, v1), vmax4(v2, v3)),
                     vmax4(vmax4(v4, v5), vmax4(v6, v7)));

        f4* op = (f4*)(out + (size_t)row * 96) + lane;
        __builtin_nontemporal_store(m, op);  // write-once output
    }
}

extern "C" void kernel_launch(void* const* d_in, const int* in_sizes, int n_in,
                              void* d_out, int out_size, void* d_ws, size_t ws_size,
                              hipStream_t stream) {
    const float* features = (const float*)d_in[0];
    const int*   in_map   = (const int*)d_in[1];
    // d_in[2] (out_map) is implied by the sorted //8 layout; d_in[3] (n_out)
    // lives on device -- derive n_out on host from out_size instead.
    float* out = (float*)d_out;

    const int C = 96;
    const int n_out = out_size / C;          // 125000
    const int rows_per_block = 8;            // 8 waves of 32 per block
    const int blocks = (n_out + rows_per_block - 1) / rows_per_block;

    SparseMaxPooling_74294344286532_kernel<<<blocks, 256, 0, stream>>>(
        features, in_map, out, n_out);
}